// CRF_54271206752592
// MI455X (gfx1250) — compile-verified
//
#include <hip/hip_runtime.h>
#include <hip/hip_bf16.h>
#include <cstdint>

typedef __attribute__((ext_vector_type(2))) float v2f;
typedef __attribute__((ext_vector_type(8))) float v8f;

constexpr int Bb = 256;
constexpr int Ss = 1024;
constexpr int Cc = 128;

// Forward Viterbi: one block per batch, 128 threads (4 wave32).
// Wave w owns output columns j in [32w, 32w+32). Transitions live in VGPRs
// in WMMA C/D layout (2 j-tiles x 8 i-blocks x 8 regs per lane).
// Score tile = alpha (x) ones + T  via  V_WMMA_F32_16X16X4_F32.
__global__ __launch_bounds__(128) void viterbi_forward(
    const float* __restrict__ emis, const int* __restrict__ mask,
    const float* __restrict__ T, uint8_t* __restrict__ bp,
    int* __restrict__ path)
{
  __shared__ float sAlpha[2][Cc];
  const int b      = blockIdx.x;
  const int tid    = threadIdx.x;
  const int w      = tid >> 5;
  const int lane   = tid & 31;
  const int laneLo = lane & 15;
  const bool hi    = lane >= 16;
  const int  hi8   = hi ? 8 : 0;
  const int  j     = (w << 5) + lane;      // this thread's output column

  // ---- preload transition tiles into registers (WMMA C/D layout) ----
  // D/C layout: lanes 0-15: M=r,   N=lane;  lanes 16-31: M=r+8, N=lane-16
  float ct[2][8][8];
#pragma unroll
  for (int tc = 0; tc < 2; ++tc)
#pragma unroll
    for (int k = 0; k < 8; ++k)
#pragma unroll
      for (int r = 0; r < 8; ++r) {
        const int row = 16 * k + r + hi8;                 // i (prev tag)
        const int col = (w << 5) + (tc << 4) + laneLo;    // j (cur tag)
        ct[tc][k][r] = T[row * Cc + col];
      }

  // B fragment: B[0,:] = 1, all other rows 0 (robust to either 4x16 layout
  // because A's K=1..3 contributions are also zeroed below).
  v2f bfrag;
  bfrag.x = hi ? 0.0f : 1.0f;
  bfrag.y = 0.0f;

  // alpha_0 = emissions[b,0,:]
  sAlpha[0][j] = emis[(size_t)b * Ss * Cc + j];
  __syncthreads();

  const float NEG = -__builtin_inff();

  for (int t = 1; t < Ss; ++t) {
    const int cur = (t - 1) & 1, nxt = t & 1;

    // issue independent global traffic early
    const float e    = emis[((size_t)b * Ss + t) * Cc + j];
    const int   m    = mask[b * Ss + t];
    const float aold = sAlpha[cur][j];
    if (t + 4 < Ss)
      __builtin_prefetch(&emis[((size_t)b * Ss + t + 4) * Cc + j], 0, 0);

    // alpha fragments for A-matrix (col K=0 only).
    // Load UNconditionally (hi lanes read the same broadcast word as their
    // low-lane partner), then zero hi lanes with a plain select so the
    // compiler emits straight-line ds_load + v_cndmask instead of
    // saveexec-branched conditional loads.
    float af[8];
#pragma unroll
    for (int k = 0; k < 8; ++k) {
      const float v = sAlpha[cur][16 * k + laneLo];
      af[k] = hi ? 0.0f : v;
    }

    // 4 independent (best,arg) chains: [j-tile][i-block parity]
    float best[2][2];
    int   arg8[2][2];
#pragma unroll
    for (int tc = 0; tc < 2; ++tc)
#pragma unroll
      for (int p = 0; p < 2; ++p) { best[tc][p] = NEG; arg8[tc][p] = 0; }

#pragma unroll
    for (int tc = 0; tc < 2; ++tc) {
#pragma unroll
      for (int k = 0; k < 8; ++k) {
        v2f a; a.x = af[k]; a.y = 0.0f;
        v8f cmat;
#pragma unroll
        for (int r = 0; r < 8; ++r) cmat[r] = ct[tc][k][r];
        // D[i,j] = alpha[i] + T[i,j]
        v8f d = __builtin_amdgcn_wmma_f32_16x16x4_f32(
            false, a, false, bfrag, (short)0, cmat, false, false);
        const int p = k & 1;
#pragma unroll
        for (int r = 0; r < 8; ++r) {
          const int  cand = k * 8 + r;          // 0..63, monotone in i per half
          const bool g    = d[r] > best[tc][p]; // strict '>' => first max kept
          best[tc][p] = g ? d[r] : best[tc][p];
          arg8[tc][p] = g ? cand : arg8[tc][p];
        }
      }
    }

    // merge parity chains, reconstruct true i, combine lane halves
    float bb[2]; int ii[2];
#pragma unroll
    for (int tc = 0; tc < 2; ++tc) {
      const bool tk1 = (best[tc][1] > best[tc][0]) ||
                       ((best[tc][1] == best[tc][0]) && (arg8[tc][1] < arg8[tc][0]));
      float bv = tk1 ? best[tc][1] : best[tc][0];
      int   a8 = tk1 ? arg8[tc][1] : arg8[tc][0];
      int   iv = a8 + ((a8 >> 3) << 3) + hi8;   // i = 16k + r (+8 for hi half)
      const float bo = __shfl_xor(bv, 16, 32);
      const int   io = __shfl_xor(iv, 16, 32);
      const bool tk2 = (bo > bv) || ((bo == bv) && (io < iv));
      bb[tc] = tk2 ? bo : bv;
      ii[tc] = tk2 ? io : iv;
    }

    const float bsel = hi ? bb[1] : bb[0];      // lane stores column j = 32w+lane
    const int   isel = hi ? ii[1] : ii[0];
    const float an   = bsel + e;
    const float afin = m ? an : aold;           // mask blend (mi in {0,1})

    sAlpha[nxt][j] = afin;
    bp[((size_t)b * (Ss - 1) + (t - 1)) * Cc + j] = (uint8_t)isel;
    __syncthreads();
  }

  // ---- best_last = argmax(final alpha), tie -> smallest index ----
  if (w == 0) {
    const float* fa = sAlpha[(Ss - 1) & 1];
    float bbest = NEG; int barg = 0;
#pragma unroll
    for (int k = 0; k < 4; ++k) {
      const int   jj = (k << 5) + lane;
      const float v  = fa[jj];
      const bool  g  = v > bbest;
      bbest = g ? v : bbest;
      barg  = g ? jj : barg;
    }
#pragma unroll
    for (int off = 16; off >= 1; off >>= 1) {
      const float bo = __shfl_xor(bbest, off, 32);
      const int   ao = __shfl_xor(barg, off, 32);
      const bool  tk = (bo > bbest) || ((bo == bbest) && (ao < barg));
      bbest = tk ? bo : bbest;
      barg  = tk ? ao : barg;
    }
    if (lane == 0) path[(size_t)b * Ss + (Ss - 1)] = barg;
  }
}

// Backtrace: one wave per batch. The bp row needed at time t is independent
// of the chased tag, so keep 4 rows in flight; tag extraction is a shuffle.
__global__ __launch_bounds__(32) void viterbi_backtrace(
    const uint8_t* __restrict__ bp, int* __restrict__ path)
{
  const int b    = blockIdx.x;
  const int lane = threadIdx.x;
  const uint8_t* bpb = bp + (size_t)b * (Ss - 1) * Cc;
  int tag = path[(size_t)b * Ss + (Ss - 1)];

  auto ROW = [&](int tt) -> unsigned {
    return ((const unsigned*)(bpb + (size_t)tt * Cc))[lane];
  };
  auto CONSUME = [&](unsigned rowv, int tt) {
    const unsigned wsel = (unsigned)__shfl((int)rowv, tag >> 2, 32);
    tag = (int)((wsel >> ((tag & 3) << 3)) & 0xFF);
    if (lane == 0) path[(size_t)b * Ss + tt] = tag;
  };

  unsigned r0 = ROW(Ss - 2), r1 = ROW(Ss - 3), r2 = ROW(Ss - 4), r3 = ROW(Ss - 5);
  for (int tb = Ss - 2; tb >= 0; tb -= 4) {
    unsigned n0 = 0, n1 = 0, n2 = 0, n3 = 0;
    if (tb - 4 >= 0) n0 = ROW(tb - 4);
    if (tb - 5 >= 0) n1 = ROW(tb - 5);
    if (tb - 6 >= 0) n2 = ROW(tb - 6);
    if (tb - 7 >= 0) n3 = ROW(tb - 7);
    CONSUME(r0, tb);
    if (tb - 1 >= 0) CONSUME(r1, tb - 1);
    if (tb - 2 >= 0) CONSUME(r2, tb - 2);
    if (tb - 3 >= 0) CONSUME(r3, tb - 3);
    r0 = n0; r1 = n1; r2 = n2; r3 = n3;
  }
}

extern "C" void kernel_launch(void* const* d_in, const int* in_sizes, int n_in,
                              void* d_out, int out_size, void* d_ws, size_t ws_size,
                              hipStream_t stream) {
  const float* emissions   = (const float*)d_in[0];   // (B,S,C) f32
  const int*   mask        = (const int*)d_in[1];     // (B,S)   i32
  const float* transitions = (const float*)d_in[2];   // (C,C)   f32
  int*         path        = (int*)d_out;             // (B,S)   i32
  uint8_t*     bp          = (uint8_t*)d_ws;          // (B,S-1,C) u8 backpointers

  viterbi_forward<<<Bb, 128, 0, stream>>>(emissions, mask, transitions, bp, path);
  viterbi_backtrace<<<Bb, 32, 0, stream>>>(bp, path);
}